// MultivariateFlowMixture_9783935500640
// MI455X (gfx1250) — compile-verified
//
#include <hip/hip_runtime.h>
#include <stdint.h>

// MultivariateFlowMixture on MI455X (gfx1250, wave32).
// Shapes fixed by harness: N=131072, V=8, K=8, B=32, L=4.
//
//  K2 (defined first so the disasm snippet shows the TDM issue):
//      4096 blocks x 256 threads; TDM (tensor_load_to_lds) stages the 103 KB
//      parameter table into LDS once per block; thread = one (n,k); 8v x 4l
//      spline evals with LDS binary search; v_rcp/v_log/v_exp transcendentals;
//      8-lane shuffle softmax; deterministic per-block partials to d_ws.
//  K1 (1 block): preprocess spline params -> table blob in d_ws.
//  K3 (1 block): deterministic tree reduction + log_softmax -> out[2].

#define TAILB 3.0f
#define ROW   33
#define OFF_CW 0
#define OFF_CH 8448
#define OFF_DV 16896
#define OFF_B  25344
#define OFF_LS 25664
#define OFF_IS 25984
#define OFF_CL 26304
#define BLOB_FLOATS 26312
#define RED_FLOATS  72        // 8 waves * 9
#define PART_OFF    26368     // blob rounded up to 64-float boundary
#define LOG2E 1.4426950408889634f
#define LN2   0.6931471805599453f

typedef unsigned int u32x4 __attribute__((ext_vector_type(4)));
typedef int          i32x4 __attribute__((ext_vector_type(4)));
typedef int          i32x8 __attribute__((ext_vector_type(8)));

__device__ __forceinline__ float fast_log(float v) {   // natural log via v_log_f32
  return LN2 * __builtin_amdgcn_logf(v);
}
__device__ __forceinline__ float fast_exp(float v) {   // exp via v_exp_f32
  return __builtin_amdgcn_exp2f(v * LOG2E);
}

// ---- Tensor Data Mover: blob (global) -> LDS, one descriptor, 1-row 2D tile ----
__device__ __forceinline__ void tdm_load_blob(const float* gsrc, uint32_t lds_off,
                                              uint32_t nf) {
  uint64_t ga = (uint64_t)(size_t)gsrc;
  // D# group 0: count=1 | lds_addr | global_addr[56:0] | type=2
  u32x4 g0 = { 1u,
               lds_off,
               (uint32_t)ga,
               (uint32_t)((ga >> 32) & 0x01FFFFFFu) | 0x80000000u };
  // D# group 1: data_size=4B; tensor_dim0=nf, tensor_dim1=1;
  //             tile_dim0=nf, tile_dim1=1, tile_dim2=0; strides=nf
  i32x8 g1 = { (int)(2u << 16),
               (int)((nf & 0xFFFFu) << 16),
               (int)(((nf >> 16) & 0xFFFFu) | (1u << 16)),
               (int)((nf & 0xFFFFu) << 16),
               1,
               (int)nf,
               (int)((nf & 0xFFFFu) << 16),
               (int)(nf >> 16) };
  i32x4 z4 = { 0, 0, 0, 0 };
#if defined(__has_include) && __has_include(<hip/amd_detail/amd_gfx1250_TDM.h>)
  i32x8 z8 = { 0, 0, 0, 0, 0, 0, 0, 0 };
  __builtin_amdgcn_tensor_load_to_lds(g0, g1, z4, z4, z8, 0);   // clang-23 arity
#else
  __builtin_amdgcn_tensor_load_to_lds(g0, g1, z4, z4, 0);       // ROCm 7.2 arity
#endif
}

// ---------------- K2: main flow evaluation (first for disasm visibility) --------
__global__ __launch_bounds__(256) void flow_main_kernel(
    const float* __restrict__ x, const float* __restrict__ blob,
    float* __restrict__ partials, int N) {
  extern __shared__ float sm[];

  // Wave 0 issues one TDM copy of the whole table into LDS (EXEC-uniform branch).
  if (threadIdx.x < 32u) {
    tdm_load_blob(blob, (uint32_t)(size_t)sm, (uint32_t)BLOB_FLOATS);
    __builtin_amdgcn_s_wait_tensorcnt(0);
  }

  int tid = blockIdx.x * 256 + threadIdx.x;
  const int total = N << 3;
  if (tid >= total) tid = total - 1;  // safety; N*8 is a multiple of 256
  const int k = tid & 7;

  // One fully-coalesced load: lane j of each 8-lane group holds x[n*8 + j].
  const float xl = x[tid];

  __syncthreads();                     // TDM data visible to all waves

  float slp = 0.f;
#pragma unroll 1
  for (int v = 0; v < 8; ++v) {
    float xv = __shfl(xl, v, 8);       // broadcast x[n*8+v] within the group
    float acc = 0.f;
    const int ab = v * 40 + k;                        // (v*5+l)*8+k
    const float* spl0 = sm + (((v * 4) * 8 + k) * ROW);
#pragma unroll
    for (int l = 0; l < 4; ++l) {
      const int a = ab + (l << 3);
      xv = (xv - sm[OFF_B + a]) * sm[OFF_IS + a];
      acc -= sm[OFF_LS + a];
      const float* CW = spl0 + l * (8 * ROW);
      const float* CH = CW + OFF_CH;
      const float* DV = CW + OFF_DV;
      const bool inside = (xv >= -TAILB) && (xv <= TAILB);
      const float xc = fminf(fmaxf(xv, -TAILB), TAILB);
      // binary search over 33 monotone knots in LDS (result in [0,31])
      int b = 0;
      b += (CW[b + 16] <= xc) ? 16 : 0;
      b += (CW[b + 8]  <= xc) ? 8  : 0;
      b += (CW[b + 4]  <= xc) ? 4  : 0;
      b += (CW[b + 2]  <= xc) ? 2  : 0;
      b += (CW[b + 1]  <= xc) ? 1  : 0;
      const float cw0 = CW[b];
      const float wdt = CW[b + 1] - cw0;
      const float ch0 = CH[b];
      const float hgt = CH[b + 1] - ch0;
      const float d0 = DV[b], d1 = DV[b + 1];
      const float invw  = __builtin_amdgcn_rcpf(wdt);
      const float delta = hgt * invw;
      const float th    = (xc - cw0) * invw;
      const float omt   = 1.f - th;
      const float t1m   = th * omt;
      const float den   = delta + (d0 + d1 - 2.f * delta) * t1m;
      const float invd  = __builtin_amdgcn_rcpf(den);
      const float y     = ch0 + hgt * (delta * th * th + d0 * t1m) * invd;
      const float dn    = (delta * delta) *
                          (d1 * th * th + 2.f * delta * t1m + d0 * omt * omt);
      const float lad   = fast_log(dn * invd * invd);  // log(dnum) - 2*log(denom)
      xv  = inside ? y : xv;
      acc += inside ? lad : 0.f;
    }
    const int a = ab + 32;                             // final affine step (l = L)
    xv = (xv - sm[OFF_B + a]) * sm[OFF_IS + a];
    acc -= sm[OFF_LS + a];
    slp += acc - 0.5f * xv * xv - 0.91893853320467274f;  // -0.5*log(2*pi)
  }

  // 8-lane softmax over k within each lane group (wave32: 4 samples / wave)
  const float cl = sm[OFF_CL + k];
  const float logit = slp + cl;
  float m = logit;
  m = fmaxf(m, __shfl_xor(m, 1, 8));
  m = fmaxf(m, __shfl_xor(m, 2, 8));
  m = fmaxf(m, __shfl_xor(m, 4, 8));
  const float e = fast_exp(logit - m);
  float se = e;
  se += __shfl_xor(se, 1, 8);
  se += __shfl_xor(se, 2, 8);
  se += __shfl_xor(se, 4, 8);
  const float p = e * __builtin_amdgcn_rcpf(se);
  float c = p * slp;                                   // contribution to log_p
  c += __shfl_xor(c, 1, 8);
  c += __shfl_xor(c, 2, 8);
  c += __shfl_xor(c, 4, 8);                            // replicated per sample

  // wave-level reduce across the 4 sample groups (xor in group-id space)
  float c2  = c + __shfl_xor(c, 8, 32);
  float cwv = c2 + __shfl_xor(c2, 16, 32);             // wave lp-sum (all lanes)
  float q2  = p + __shfl_xor(p, 8, 32);
  float qwv = q2 + __shfl_xor(q2, 16, 32);             // lane i: wave q-sum, k=i&7

  float* wred = sm + BLOB_FLOATS;
  const int wave = threadIdx.x >> 5;
  const int lane = threadIdx.x & 31;
  if (lane == 0) wred[wave * 9 + 0] = cwv;
  if (lane < 8)  wred[wave * 9 + 1 + lane] = qwv;
  __syncthreads();
  if (threadIdx.x < 9) {                               // deterministic block partial
    float sacc = 0.f;
#pragma unroll
    for (int wv = 0; wv < 8; ++wv) sacc += wred[wv * 9 + threadIdx.x];
    partials[(size_t)blockIdx.x * 9 + threadIdx.x] = sacc;
  }
}

// ---------------- K1: parameter preprocessing (tiny) ----------------
__global__ __launch_bounds__(256) void flow_precompute_kernel(
    const float* __restrict__ w, const float* __restrict__ h,
    const float* __restrict__ s, const float* __restrict__ bias,
    const float* __restrict__ ls, const float* __restrict__ cl,
    float* __restrict__ blob) {
  const int t = threadIdx.x;  // spline id: t = ((v*4+l)*8+k), 256 splines
  // widths -> cumw   (widths = 1e-3 + (1-32e-3)*softmax; cumw = 6*cumsum-3)
  {
    const float* u = w + t * 32;
    float m = u[0];
    for (int b = 1; b < 32; ++b) m = fmaxf(m, u[b]);
    float sum = 0.f;
    for (int b = 0; b < 32; ++b) sum += expf(u[b] - m);
    float inv = 1.f / sum;
    float cum = 0.f;
    float* CW = blob + OFF_CW + t * ROW;
    CW[0] = -TAILB;
    for (int b = 0; b < 32; ++b) {
      cum += 1e-3f + 0.968f * expf(u[b] - m) * inv;
      CW[b + 1] = (b == 31) ? TAILB : (2.f * TAILB * cum - TAILB);
    }
  }
  // heights -> cumh
  {
    const float* u = h + t * 32;
    float m = u[0];
    for (int b = 1; b < 32; ++b) m = fmaxf(m, u[b]);
    float sum = 0.f;
    for (int b = 0; b < 32; ++b) sum += expf(u[b] - m);
    float inv = 1.f / sum;
    float cum = 0.f;
    float* CH = blob + OFF_CH + t * ROW;
    CH[0] = -TAILB;
    for (int b = 0; b < 32; ++b) {
      cum += 1e-3f + 0.968f * expf(u[b] - m) * inv;
      CH[b + 1] = (b == 31) ? TAILB : (2.f * TAILB * cum - TAILB);
    }
  }
  // derivatives: boundary softplus(log(expm1(1-MIN_D))) -> exactly 1.0
  {
    float* D = blob + OFF_DV + t * ROW;
    D[0] = 1.f;
    D[32] = 1.f;
    const float* u = s + t * 31;
    for (int j = 0; j < 31; ++j) {
      float v = u[j];
      float sp = (v > 20.f) ? v : log1pf(expf(v));
      D[j + 1] = 1e-3f + sp;
    }
  }
  // affine params [V][L+1][K] = 320 each
  for (int i = t; i < 320; i += 256) {
    blob[OFF_B + i] = bias[i];
    float L = ls[i];
    blob[OFF_LS + i] = L;
    blob[OFF_IS + i] = expf(-L);
  }
  if (t < 8) blob[OFF_CL + t] = cl[t];
}

// ---------------- K3: deterministic finalize ----------------
__global__ __launch_bounds__(256) void flow_finalize_kernel(
    const float* __restrict__ partials, int nblocks,
    const float* __restrict__ comp_logits, float* __restrict__ out, int N) {
  __shared__ float red[256];
  __shared__ float res[9];
  for (int j = 0; j < 9; ++j) {
    float s = 0.f;
    for (int i = threadIdx.x; i < nblocks; i += 256) s += partials[(size_t)i * 9 + j];
    red[threadIdx.x] = s;
    __syncthreads();
    for (int o = 128; o > 0; o >>= 1) {
      if ((int)threadIdx.x < o) red[threadIdx.x] += red[threadIdx.x + o];
      __syncthreads();
    }
    if (threadIdx.x == 0) res[j] = red[0];
    __syncthreads();
  }
  if (threadIdx.x == 0) {
    const float invN = 1.f / (float)N;
    float m = comp_logits[0];
    for (int q = 1; q < 8; ++q) m = fmaxf(m, comp_logits[q]);
    float se = 0.f;
    for (int q = 0; q < 8; ++q) se += expf(comp_logits[q] - m);
    const float lse = logf(se) + m;
    float closs = 0.f;
    for (int q = 0; q < 8; ++q)
      closs -= (res[1 + q] * invN) * (comp_logits[q] - lse);
    out[0] = res[0] * invN;   // log_p
    out[1] = closs;           // comp_loss
  }
}

extern "C" void kernel_launch(void* const* d_in, const int* in_sizes, int n_in,
                              void* d_out, int out_size, void* d_ws, size_t ws_size,
                              hipStream_t stream) {
  (void)n_in; (void)out_size; (void)ws_size;
  const float* x    = (const float*)d_in[0];
  const float* w    = (const float*)d_in[1];
  const float* h    = (const float*)d_in[2];
  const float* s    = (const float*)d_in[3];
  const float* bias = (const float*)d_in[4];
  const float* ls   = (const float*)d_in[5];
  const float* cl   = (const float*)d_in[6];
  float* out = (float*)d_out;

  const int N = in_sizes[0] / 8;                  // 131072
  float* blob     = (float*)d_ws;                 // BLOB_FLOATS floats
  float* partials = blob + PART_OFF;              // nb*9 floats

  const int total = N * 8;
  const int nb = (total + 255) / 256;             // 4096

  flow_precompute_kernel<<<1, 256, 0, stream>>>(w, h, s, bias, ls, cl, blob);
  flow_main_kernel<<<nb, 256, (BLOB_FLOATS + RED_FLOATS) * sizeof(float), stream>>>(
      x, blob, partials, N);
  flow_finalize_kernel<<<1, 256, 0, stream>>>(partials, nb, cl, out, N);
}